// Feat2Graph_9835475108512
// MI455X (gfx1250) — compile-verified
//
#include <hip/hip_runtime.h>
#include <hip/hip_bf16.h>

typedef float v2f __attribute__((ext_vector_type(2)));
typedef float v8f __attribute__((ext_vector_type(8)));

#define FDIM 256
#define NSEQ 4096
#define NBAT 4

__device__ __forceinline__ v8f wmma_f32_k4(v2f a, v2f b, v8f c) {
  // D = A(16x4,f32) * B(4x16,f32) + C(16x16,f32)
  return __builtin_amdgcn_wmma_f32_16x16x4_f32(
      /*neg_a=*/false, a, /*neg_b=*/false, b,
      /*c_mod=*/(short)0, c, /*reuse_a=*/false, /*reuse_b=*/false);
}

// ---------------------------------------------------------------------------
// Kernel 1: qx = x @ Wq^T + bq ; kx = x @ Wk^T + bk
// grid = (1024, 2), block = 256 (8 waves). Each wave: one 16x16 tile of BOTH
// qx and kx (A fragment of x feeds both WMMAs per K-step).
// Software-pipelined: fragments for step k+4 are issued before step k's WMMAs.
// ---------------------------------------------------------------------------
__global__ void proj_kernel(const float* __restrict__ x,
                            const float* __restrict__ Wq, const float* __restrict__ bq,
                            const float* __restrict__ Wk, const float* __restrict__ bk,
                            float* __restrict__ qx, float* __restrict__ kx) {
  const int wave = threadIdx.x >> 5;
  const int lane = threadIdx.x & 31;
  const int half = lane >> 4;        // 0: lanes 0-15, 1: lanes 16-31
  const int lid  = lane & 15;

  const int row0 = blockIdx.x << 4;                 // 0..16383 step 16
  const int ct   = (blockIdx.y << 3) + wave;        // col tile 0..15
  const int col0 = ct << 4;

  const float* __restrict__ xrow  = x  + (size_t)(row0 + lid) * FDIM + 2 * half;
  const float* __restrict__ wqrow = Wq + (size_t)(col0 + lid) * FDIM + 2 * half;
  const float* __restrict__ wkrow = Wk + (size_t)(col0 + lid) * FDIM + 2 * half;

  v8f cq = {};
  v8f ck = {};

  // prologue: fragments for k = 0
  v2f a  = *(const v2f*)(xrow);
  v2f b0 = *(const v2f*)(wqrow);
  v2f b1 = *(const v2f*)(wkrow);

#pragma unroll 4
  for (int k = 4; k < FDIM; k += 4) {
    // issue next iteration's loads first
    v2f an  = *(const v2f*)(xrow  + k);
    v2f b0n = *(const v2f*)(wqrow + k);
    v2f b1n = *(const v2f*)(wkrow + k);
    // consume current fragments
    cq = wmma_f32_k4(a, b0, cq);
    ck = wmma_f32_k4(a, b1, ck);
    a = an; b0 = b0n; b1 = b1n;
  }
  // epilogue
  cq = wmma_f32_k4(a, b0, cq);
  ck = wmma_f32_k4(a, b1, ck);

  const float biasq = bq[col0 + lid];
  const float biask = bk[col0 + lid];
#pragma unroll
  for (int r = 0; r < 8; ++r) {
    const size_t row = (size_t)(row0 + r + 8 * half);
    qx[row * FDIM + col0 + lid] = cq[r] + biasq;
    kx[row * FDIM + col0 + lid] = ck[r] + biask;
  }
}

// ---------------------------------------------------------------------------
// Kernel 2: sq[b,i,j] = (qx[b,i,:] . kx[b,j,:])^2   (unnormalized)
// grid = (8, 256, 4), block = 256 (8 waves). Each wave: a 16x64 strip
// (4 column tiles). One A fragment feeds 4 WMMAs per K-step.
// Software-pipelined K-loop (next step's 5 fragments in flight during WMMAs).
// ---------------------------------------------------------------------------
__global__ void dotsq_kernel(const float* __restrict__ qx,
                             const float* __restrict__ kx,
                             float* __restrict__ adj) {
  const int b    = blockIdx.z;
  const int i0   = blockIdx.y << 4;                 // row tile
  const int wave = threadIdx.x >> 5;
  const int lane = threadIdx.x & 31;
  const int half = lane >> 4;
  const int lid  = lane & 15;

  const int strip = (blockIdx.x << 3) + wave;       // 0..63
  const int j0    = strip << 6;                     // 64 cols per wave

  const float* __restrict__ qrow = qx + ((size_t)b * NSEQ + i0 + lid) * FDIM + 2 * half;
  const float* __restrict__ krow = kx + ((size_t)b * NSEQ + j0 + lid) * FDIM + 2 * half;

  v8f c0 = {}, c1 = {}, c2 = {}, c3 = {};

  // prologue: fragments for k = 0
  v2f a  = *(const v2f*)(qrow);
  v2f b0 = *(const v2f*)(krow);
  v2f b1 = *(const v2f*)(krow + 16 * FDIM);
  v2f b2 = *(const v2f*)(krow + 32 * FDIM);
  v2f b3 = *(const v2f*)(krow + 48 * FDIM);

#pragma unroll 4
  for (int k = 4; k < FDIM; k += 4) {
    // issue next iteration's loads first
    v2f an  = *(const v2f*)(qrow + k);
    v2f b0n = *(const v2f*)(krow + k);
    v2f b1n = *(const v2f*)(krow + 16 * FDIM + k);
    v2f b2n = *(const v2f*)(krow + 32 * FDIM + k);
    v2f b3n = *(const v2f*)(krow + 48 * FDIM + k);
    // consume current fragments
    c0 = wmma_f32_k4(a, b0, c0);
    c1 = wmma_f32_k4(a, b1, c1);
    c2 = wmma_f32_k4(a, b2, c2);
    c3 = wmma_f32_k4(a, b3, c3);
    a = an; b0 = b0n; b1 = b1n; b2 = b2n; b3 = b3n;
  }
  // epilogue
  c0 = wmma_f32_k4(a, b0, c0);
  c1 = wmma_f32_k4(a, b1, c1);
  c2 = wmma_f32_k4(a, b2, c2);
  c3 = wmma_f32_k4(a, b3, c3);

  // out element (row = i0 + r + 8*half, col = j0 + 16*t + lid)
  float* __restrict__ out =
      adj + ((size_t)b * NSEQ + i0 + 8 * half) * NSEQ + j0 + lid;
#pragma unroll
  for (int r = 0; r < 8; ++r) {
    const size_t ro = (size_t)r * NSEQ;
    float v0 = c0[r], v1 = c1[r], v2 = c2[r], v3 = c3[r];
    out[ro +  0] = v0 * v0;
    out[ro + 16] = v1 * v1;
    out[ro + 32] = v2 * v2;
    out[ro + 48] = v3 * v3;
  }
}

// ---------------------------------------------------------------------------
// Kernel 3: L1-normalize each adj row in place; also copy x into d_out.
// grid = 16384 (one block per global row), block = 256.
// ---------------------------------------------------------------------------
__global__ void normrow_kernel(const float* __restrict__ x,
                               float* __restrict__ out_x,
                               float* __restrict__ adj) {
  const size_t r  = blockIdx.x;                     // 0..16383
  const int   tid = threadIdx.x;

  // x passthrough: block r copies x[r*256 .. r*256+255]
  out_x[r * FDIM + tid] = x[r * FDIM + tid];

  float4* __restrict__ row4 = (float4*)(adj + r * (size_t)NSEQ);

  float4 v[4];
  float s = 0.0f;
#pragma unroll
  for (int i = 0; i < 4; ++i) {
    v[i] = row4[tid + 256 * i];
    s += v[i].x + v[i].y + v[i].z + v[i].w;
  }

  __shared__ float red[256];
  red[tid] = s;
  __syncthreads();
#pragma unroll
  for (int off = 128; off > 0; off >>= 1) {
    if (tid < off) red[tid] += red[tid + off];
    __syncthreads();
  }
  const float denom = fmaxf(red[0], 1e-12f);
  const float inv = 1.0f / denom;

#pragma unroll
  for (int i = 0; i < 4; ++i) {
    float4 w = v[i];
    w.x *= inv; w.y *= inv; w.z *= inv; w.w *= inv;
    row4[tid + 256 * i] = w;
  }
}

// ---------------------------------------------------------------------------
extern "C" void kernel_launch(void* const* d_in, const int* in_sizes, int n_in,
                              void* d_out, int out_size, void* d_ws, size_t ws_size,
                              hipStream_t stream) {
  const float* x  = (const float*)d_in[0];
  const float* Wq = (const float*)d_in[1];
  const float* bq = (const float*)d_in[2];
  const float* Wk = (const float*)d_in[3];
  const float* bk = (const float*)d_in[4];

  float* out   = (float*)d_out;
  float* out_x = out;                                      // [4*4096*256]
  float* adj   = out + (size_t)NBAT * NSEQ * FDIM;         // [4*4096*4096]

  float* qx = (float*)d_ws;                                // [16384*256]
  float* kx = qx + (size_t)NBAT * NSEQ * FDIM;             // [16384*256]

  dim3 g1(1024, 2, 1);
  proj_kernel<<<g1, 256, 0, stream>>>(x, Wq, bq, Wk, bk, qx, kx);

  dim3 g2(8, 256, NBAT);
  dotsq_kernel<<<g2, 256, 0, stream>>>(qx, kx, adj);

  normrow_kernel<<<16384, 256, 0, stream>>>(x, out_x, adj);
}